// ChamferDistance_38843684225041
// MI455X (gfx1250) — compile-verified
//
#include <hip/hip_runtime.h>
#include <math.h>

// CDNA5 wave32 WMMA fragments for V_WMMA_F32_16X16X4_F32:
//   A: 16x4 f32  -> 2 VGPRs/lane (v2f)   VGPR0 = K0 (lanes 0-15)/K2 (lanes 16-31); VGPR1 = K1/K3
//   B: 4x16 f32  -> 2 VGPRs/lane (v2f)   same K split as A
//   C/D: 16x16 f32 -> 8 VGPRs/lane (v8f) VGPR r = row r (lanes 0-15) / row r+8 (lanes 16-31)
typedef __attribute__((ext_vector_type(2))) float v2f;
typedef __attribute__((ext_vector_type(8))) float v8f;

#define WAVES_PER_BLOCK 4

// ---------------------------------------------------------------------------
// Prep: build the wave32 B-matrix fragment for each 16-point tile of Y.
//   B col j = (y0, y1, y2, |y_j|^2); lane layout: .x = K0 (lanes 0-15)/K2 (16-31),
//   .y = K1/K3. Stored as [batch][tile][lane] float2 (256 B per tile, coalesced).
// Point index is clamped to M-1, so tail tiles duplicate real points (min-safe).
// ---------------------------------------------------------------------------
__global__ __launch_bounds__(256) void build_yfrags(
    const float* __restrict__ Y, float* __restrict__ Yf, int M, int ntiles)
{
    const int t = blockIdx.x * blockDim.x + threadIdx.x;   // (tile,lane) within batch
    if (t >= ntiles * 32) return;
    const int tile = t >> 5;
    const int lane = t & 31;
    const int l16  = lane & 15;
    const bool hi  = lane >= 16;

    const float* Yb = Y + (size_t)blockIdx.y * M * 3;
    const int j = min(tile * 16 + l16, M - 1);
    const float y0 = Yb[j * 3 + 0];
    const float y1 = Yb[j * 3 + 1];
    const float y2 = Yb[j * 3 + 2];
    const float yn = y0 * y0 + y1 * y1 + y2 * y2;

    float* out = Yf + ((size_t)blockIdx.y * ntiles + tile) * 64 + lane * 2;
    out[0] = hi ? y2 : y0;
    out[1] = hi ? yn : y1;
}

// ---------------------------------------------------------------------------
// Main pass: each wave owns 32 x-rows (two A frags + two loop-invariant C frags)
// and streams prebuilt B fragments; 2 WMMAs per tile:
//   A row i = (-2x0, -2x1, -2x2, 1),  C[i][j] = |x_i|^2
//   => D[i][j] = |x_i|^2 - 2 x_i.y_j + |y_j|^2
// Inner loop = 1 coalesced b64 frag load + 2 wmma + 8 v_min3 per tile.
// ---------------------------------------------------------------------------
__global__ __launch_bounds__(32 * WAVES_PER_BLOCK) void chamfer_nn_rows(
    const float* __restrict__ X, const float* __restrict__ Yf,
    float* __restrict__ outmin, int N, int ntiles)
{
    const int lane = threadIdx.x & 31;
    const int wave = threadIdx.x >> 5;
    const int strip = blockIdx.x * WAVES_PER_BLOCK + wave;
    const int row_base = strip * 32;
    if (row_base >= N) return;                 // wave-uniform: EXEC stays all-1s for WMMA
    const int b = blockIdx.y;

    const float* Xb  = X + (size_t)b * N * 3;
    const float* Yfb = Yf + (size_t)b * ntiles * 64;
    float* outb = outmin + (size_t)b * N;

    const int  l16 = lane & 15;
    const bool hi  = lane >= 16;

    // Two 16-row strips: rows row_base+l16 and row_base+16+l16 (clamped for safety).
    const int r0 = min(row_base + l16,      N - 1);
    const int r1 = min(row_base + 16 + l16, N - 1);
    const float x00 = Xb[r0 * 3 + 0], x01 = Xb[r0 * 3 + 1], x02 = Xb[r0 * 3 + 2];
    const float x10 = Xb[r1 * 3 + 0], x11 = Xb[r1 * 3 + 1], x12 = Xb[r1 * 3 + 2];
    const float xn0 = x00 * x00 + x01 * x01 + x02 * x02;
    const float xn1 = x10 * x10 + x11 * x11 + x12 * x12;

    // A frags with the -2 folded in (loop-invariant).
    v2f a0, a1;
    a0.x = hi ? (-2.0f * x02) : (-2.0f * x00);
    a0.y = hi ? 1.0f          : (-2.0f * x01);
    a1.x = hi ? (-2.0f * x12) : (-2.0f * x10);
    a1.y = hi ? 1.0f          : (-2.0f * x11);

    // C frags: VGPR r needs |x_row|^2 for row r (lanes 0-15) / r+8 (lanes 16-31);
    // lane r holds xn for row row_base+r -> broadcast from lane (r + (hi?8:0)).
    const int srcofs = hi ? 8 : 0;
    v8f c0, c1;
#pragma unroll
    for (int r = 0; r < 8; ++r) {
        c0[r] = __shfl(xn0, r + srcofs, 32);
        c1[r] = __shfl(xn1, r + srcofs, 32);
    }

    float rmin0[8], rmin1[8];
#pragma unroll
    for (int r = 0; r < 8; ++r) { rmin0[r] = 3.0e38f; rmin1[r] = 3.0e38f; }

    const float* fragp = Yfb + lane * 2;
#pragma unroll 4
    for (int tile = 0; tile < ntiles; ++tile) {
        const v2f bf = *(const v2f*)(fragp + (size_t)tile * 64);

        const v8f d0 = __builtin_amdgcn_wmma_f32_16x16x4_f32(
            false, a0, false, bf, (short)0, c0, false, false);
        const v8f d1 = __builtin_amdgcn_wmma_f32_16x16x4_f32(
            false, a1, false, bf, (short)0, c1, false, false);

#pragma unroll
        for (int r = 0; r < 8; ++r) {
            rmin0[r] = fminf(rmin0[r], d0[r]);
            rmin1[r] = fminf(rmin1[r], d1[r]);
        }
    }

    // Reduce across the 16-lane halves (masks <16 never cross halves), then clamp
    // (min_j max(d,0) == max(min_j d, 0), hoisted out of the loop).
#pragma unroll
    for (int r = 0; r < 8; ++r) {
        float v0 = rmin0[r], v1 = rmin1[r];
        v0 = fminf(v0, __shfl_xor(v0, 1, 32));
        v1 = fminf(v1, __shfl_xor(v1, 1, 32));
        v0 = fminf(v0, __shfl_xor(v0, 2, 32));
        v1 = fminf(v1, __shfl_xor(v1, 2, 32));
        v0 = fminf(v0, __shfl_xor(v0, 4, 32));
        v1 = fminf(v1, __shfl_xor(v1, 4, 32));
        v0 = fminf(v0, __shfl_xor(v0, 8, 32));
        v1 = fminf(v1, __shfl_xor(v1, 8, 32));
        rmin0[r] = fmaxf(v0, 0.0f);
        rmin1[r] = fmaxf(v1, 0.0f);
    }

    // Row r of set0 lives in lanes 0-15 (r+8 in lanes 16-31); set1 is rows +16.
    if (lane == 0) {
#pragma unroll
        for (int r = 0; r < 8; ++r) {
            if (row_base + r      < N) outb[row_base + r]      = rmin0[r];
            if (row_base + 16 + r < N) outb[row_base + 16 + r] = rmin1[r];
        }
    } else if (lane == 16) {
#pragma unroll
        for (int r = 0; r < 8; ++r) {
            if (row_base + 8 + r  < N) outb[row_base + 8 + r]  = rmin0[r];
            if (row_base + 24 + r < N) outb[row_base + 24 + r] = rmin1[r];
        }
    }
}

// out = 0.5 * (mean(sqrt(m1)) + mean(sqrt(m2))). 64K elements total: one block.
__global__ __launch_bounds__(1024) void reduce_mean_sqrt(
    const float* __restrict__ m1, int n1,
    const float* __restrict__ m2, int n2,
    float* __restrict__ out)
{
    __shared__ float sm1[32], sm2[32];
    float s1 = 0.0f, s2 = 0.0f;
    for (int i = threadIdx.x; i < n1; i += blockDim.x) s1 += sqrtf(m1[i]);
    for (int i = threadIdx.x; i < n2; i += blockDim.x) s2 += sqrtf(m2[i]);
#pragma unroll
    for (int m = 16; m >= 1; m >>= 1) {
        s1 += __shfl_xor(s1, m, 32);
        s2 += __shfl_xor(s2, m, 32);
    }
    const int wid = threadIdx.x >> 5;
    if ((threadIdx.x & 31) == 0) { sm1[wid] = s1; sm2[wid] = s2; }
    __syncthreads();
    if (threadIdx.x < 32) {
        const int nw = blockDim.x >> 5;
        s1 = (threadIdx.x < nw) ? sm1[threadIdx.x] : 0.0f;
        s2 = (threadIdx.x < nw) ? sm2[threadIdx.x] : 0.0f;
#pragma unroll
        for (int m = 16; m >= 1; m >>= 1) {
            s1 += __shfl_xor(s1, m, 32);
            s2 += __shfl_xor(s2, m, 32);
        }
        if (threadIdx.x == 0)
            out[0] = 0.5f * (s1 / (float)n1 + s2 / (float)n2);
    }
}

extern "C" void kernel_launch(void* const* d_in, const int* in_sizes, int n_in,
                              void* d_out, int out_size, void* d_ws, size_t ws_size,
                              hipStream_t stream) {
    const float* pcs1 = (const float*)d_in[0];   // (B,N,3) fp32
    const float* pcs2 = (const float*)d_in[1];   // (B,M,3) fp32
    const int B = 4;
    const int N = in_sizes[0] / (B * 3);
    const int M = in_sizes[1] / (B * 3);
    const int nt2 = (M + 15) / 16;               // tiles over pcs2 (pass 1 columns)
    const int nt1 = (N + 15) / 16;               // tiles over pcs1 (pass 2 columns)

    // Workspace layout (floats): Yf2 | Yf1 | min1 | min2  (~1.25 MB for this shape)
    float* Yf2  = (float*)d_ws;                  // B * nt2 * 64
    float* Yf1  = Yf2 + (size_t)B * nt2 * 64;    // B * nt1 * 64
    float* min1 = Yf1 + (size_t)B * nt1 * 64;    // B * N
    float* min2 = min1 + (size_t)B * N;          // B * M

    // Build B-matrix fragments for both point clouds.
    build_yfrags<<<dim3((nt2 * 32 + 255) / 256, B), 256, 0, stream>>>(pcs2, Yf2, M, nt2);
    build_yfrags<<<dim3((nt1 * 32 + 255) / 256, B), 256, 0, stream>>>(pcs1, Yf1, N, nt1);

    dim3 blk(32 * WAVES_PER_BLOCK);
    dim3 g1((N / 32 + WAVES_PER_BLOCK - 1) / WAVES_PER_BLOCK, B);
    dim3 g2((M / 32 + WAVES_PER_BLOCK - 1) / WAVES_PER_BLOCK, B);

    // Pass 1: NN(x -> y). Pass 2: NN(y -> x). No atomics: deterministic.
    chamfer_nn_rows<<<g1, blk, 0, stream>>>(pcs1, Yf2, min1, N, nt2);
    chamfer_nn_rows<<<g2, blk, 0, stream>>>(pcs2, Yf1, min2, M, nt1);
    reduce_mean_sqrt<<<1, 1024, 0, stream>>>(min1, B * N, min2, B * M, (float*)d_out);
}